// ThreeNNLFrames_32006096289979
// MI455X (gfx1250) — compile-verified
//
#include <hip/hip_runtime.h>
#include <math.h>

// 3-NN + local frames for N=16384 points in R^3 on gfx1250 (MI455X).
//
// d2[i][j] = |pi|^2 + |pj|^2 - 2*(pos . pos^T). The K=3 GEMM maps onto
// V_WMMA_F32_16X16X4_F32 (K padded to 4 with zeros). One wave owns a 16-row
// strip and sweeps all N/16 column tiles (one WMMA each). Top-3 selection is
// done on packed u64 keys (f32 distance bits << 32 | column index) with a
// branch-free min/max sorting network (lowers to native v_min_u64/v_max_u64,
// dual-issued alongside the WMMA). Load latency in the tile loop is hidden by
// wave-level parallelism (8 waves/block, 1024 waves, cache-resident input) —
// manual software pipelining was tried and produced worse codegen (the
// compiler rotates it away and splits the b96 load). Cross-lane merge via a
// per-wave LDS region, then Gram-Schmidt (norm + 1e-12, matching the JAX
// reference) on lanes 0..15.

typedef float v2f __attribute__((ext_vector_type(2)));
typedef float v8f __attribute__((ext_vector_type(8)));

#define WAVES_PER_BLOCK 8
#define BLOCK_THREADS (WAVES_PER_BLOCK * 32)

typedef unsigned long long u64;
typedef unsigned int u32;

// Insert key e into sorted triple k0 <= k1 <= k2 (keep 3 smallest).
// min/max pairs -> llvm.umin/umax -> v_min_u64 / v_max_u64, no control flow.
__device__ __forceinline__ void kmin3_insert(u64 e, u64& k0, u64& k1, u64& k2) {
  u64 lo0 = (e < k0) ? e : k0;
  u64 hi0 = (e < k0) ? k0 : e;
  u64 lo1 = (hi0 < k1) ? hi0 : k1;
  u64 hi1 = (hi0 < k1) ? k1 : hi0;
  u64 lo2 = (hi1 < k2) ? hi1 : k2;
  k0 = lo0; k1 = lo1; k2 = lo2;
}

__launch_bounds__(BLOCK_THREADS, 2)
__global__ void ThreeNNLFrames_kernel(const float* __restrict__ pos,
                                      float* __restrict__ out, int n) {
  const int lane = threadIdx.x & 31;
  const int w    = threadIdx.x >> 5;
  const int hi   = lane >> 4;    // 0: lanes 0-15, 1: lanes 16-31
  const int ln   = lane & 15;
  const int strip = blockIdx.x * WAVES_PER_BLOCK + w;
  const int m0 = strip * 16;

  // Per-wave private merge region: 16 rows x (16 lanes x 3 packed candidates).
  __shared__ u64 lds_k[WAVES_PER_BLOCK][16][48];

  const bool active = (m0 < n);

  // -------- A operand: rows m0..m0+15 ------------------------------------
  // ISA 7.12.2 "32-bit A-Matrix 16x4": lanes 0-15 hold K=0,1; lanes 16-31
  // hold K=2,3 for the same M=0..15. K=3 column is the zero pad.
  v2f a;
  a.x = 0.0f; a.y = 0.0f;
  float sqm[8];
#pragma unroll
  for (int r = 0; r < 8; ++r) sqm[r] = 0.0f;
  if (active) {
    int mrow = m0 + ln;
    if (mrow >= n) mrow = n - 1;
    const float ax = pos[3 * mrow + 0];
    const float ay = pos[3 * mrow + 1];
    const float az = pos[3 * mrow + 2];
    a.x = hi ? az : ax;
    a.y = hi ? 0.0f : ay;
    // |p_m|^2 for the 8 C-rows this lane owns: m = m0 + r + 8*hi.
#pragma unroll
    for (int r = 0; r < 8; ++r) {
      int mr = m0 + r + hi * 8;
      if (mr >= n) mr = n - 1;
      const float x = pos[3 * mr + 0];
      const float y = pos[3 * mr + 1];
      const float z = pos[3 * mr + 2];
      sqm[r] = x * x + y * y + z * z;
    }
  }

  // -------- per-lane top-3 keys for 8 rows --------
  u64 k0[8], k1[8], k2[8];
#pragma unroll
  for (int r = 0; r < 8; ++r) {
    k0[r] = ~0ull; k1[r] = ~0ull; k2[r] = ~0ull;
  }

  if (active) {
    const int ntiles = n >> 4;
    for (int t = 0; t < ntiles; ++t) {
      const int ncol = (t << 4) + ln;          // column this lane owns
      // B operand (4x16, K-major like A): lanes 0-15 -> K=0,1; 16-31 -> K=2,pad
      const float px = pos[3 * ncol + 0];
      const float py = pos[3 * ncol + 1];
      const float pz = pos[3 * ncol + 2];
      v2f b;
      b.x = hi ? pz : px;
      b.y = hi ? 0.0f : py;
      const float sqn = px * px + py * py + pz * pz;

      v8f c = {};                              // inline-zero accumulator
      c = __builtin_amdgcn_wmma_f32_16x16x4_f32(
          /*neg_a=*/false, a, /*neg_b=*/false, b,
          /*c_mod=*/(short)0, c, /*reuse_a=*/false, /*reuse_b=*/false);

      // Which unrolled row index hits the diagonal in this tile (if any).
      const int rdiag = ncol - m0 - hi * 8;    // r == rdiag  <=>  m == ncol

#pragma unroll
      for (int r = 0; r < 8; ++r) {
        float d = sqm[r] + sqn - 2.0f * c[r];
        d = fmaxf(d, 0.0f);                    // fp guard: keep bits ordered
        u32 db = __float_as_uint(d);
        db = (r == rdiag) ? 0x7f800000u : db;  // +inf bits on self-loop
        const u64 key = ((u64)db << 32) | (u32)ncol;
        kmin3_insert(key, k0[r], k1[r], k2[r]);
      }
    }

    // -------- dump per-lane candidates to this wave's LDS region --------
#pragma unroll
    for (int r = 0; r < 8; ++r) {
      const int row = r + hi * 8;
      lds_k[w][row][ln * 3 + 0] = k0[r];
      lds_k[w][row][ln * 3 + 1] = k1[r];
      lds_k[w][row][ln * 3 + 2] = k2[r];
    }
  }

  __syncthreads();

  // -------- cross-lane merge + Gram-Schmidt: lanes 0..15, one row each ----
  if (active && lane < 16) {
    const int m = m0 + lane;
    if (m < n) {
      u64 b0 = ~0ull, b1 = ~0ull, b2 = ~0ull;
#pragma unroll 4
      for (int s = 0; s < 48; ++s) {
        kmin3_insert(lds_k[w][lane][s], b0, b1, b2);
      }
      const int i0 = (int)(u32)b0;
      const int i1 = (int)(u32)b1;
      const int i2 = (int)(u32)b2;

      const float px = pos[3 * m + 0], py = pos[3 * m + 1], pz = pos[3 * m + 2];
      float xx = pos[3 * i0 + 0] - px, xy = pos[3 * i0 + 1] - py, xz = pos[3 * i0 + 2] - pz;
      float yx = pos[3 * i1 + 0] - px, yy = pos[3 * i1 + 1] - py, yz = pos[3 * i1 + 2] - pz;
      float zx = pos[3 * i2 + 0] - px, zy = pos[3 * i2 + 1] - py, zz = pos[3 * i2 + 2] - pz;

      const float EPS = 1e-12f;
      // e1 = normalize(x)
      float nx = sqrtf(xx * xx + xy * xy + xz * xz) + EPS;
      float e1x = xx / nx, e1y = xy / nx, e1z = xz / nx;
      // e2 = normalize(y - (y.e1) e1)
      float dy1 = yx * e1x + yy * e1y + yz * e1z;
      float ypx = yx - dy1 * e1x, ypy = yy - dy1 * e1y, ypz = yz - dy1 * e1z;
      float ny = sqrtf(ypx * ypx + ypy * ypy + ypz * ypz) + EPS;
      float e2x = ypx / ny, e2y = ypy / ny, e2z = ypz / ny;
      // e3 = normalize(z - (z.e1) e1 - (z.e2) e2)
      float dz1 = zx * e1x + zy * e1y + zz * e1z;
      float dz2 = zx * e2x + zy * e2y + zz * e2z;
      float zpx = zx - dz1 * e1x - dz2 * e2x;
      float zpy = zy - dz1 * e1y - dz2 * e2y;
      float zpz = zz - dz1 * e1z - dz2 * e2z;
      float nz = sqrtf(zpx * zpx + zpy * zpy + zpz * zpz) + EPS;
      float e3x = zpx / nz, e3y = zpy / nz, e3z = zpz / nz;

      float* o = out + (size_t)m * 9;
      o[0] = e1x; o[1] = e1y; o[2] = e1z;
      o[3] = e2x; o[4] = e2y; o[5] = e2z;
      o[6] = e3x; o[7] = e3y; o[8] = e3z;
    }
  }
}

extern "C" void kernel_launch(void* const* d_in, const int* in_sizes, int n_in,
                              void* d_out, int out_size, void* d_ws, size_t ws_size,
                              hipStream_t stream) {
  (void)n_in; (void)d_ws; (void)ws_size; (void)out_size;
  const float* pos = (const float*)d_in[0];
  float* out = (float*)d_out;
  const int n = in_sizes[0] / 3;                 // 16384 points
  const int strips = (n + 15) / 16;              // 16-row strips, 1 wave each
  const int blocks = (strips + WAVES_PER_BLOCK - 1) / WAVES_PER_BLOCK;
  ThreeNNLFrames_kernel<<<blocks, BLOCK_THREADS, 0, stream>>>(pos, out, n);
}